// TFN2FrameCrossIPA_56023553409099
// MI455X (gfx1250) — compile-verified
//
#include <hip/hip_runtime.h>
#include <math.h>

typedef __attribute__((ext_vector_type(2))) float v2f;
typedef __attribute__((ext_vector_type(8))) float v8f;

#define CS 384
#define CZ 128
#define CH_ 16
#define H_ 8
#define PQK_ 8
#define FS_ 128
#define PIN_ 16
#define HC_ 128
#define OUT_IN_ 416
#define KC_ 128  // K-chunk staged in LDS

// ---------------- order-preserving float<->uint for atomic max ----------------
__device__ __forceinline__ unsigned enc_f(float f) {
  unsigned u = __float_as_uint(f);
  return (u & 0x80000000u) ? ~u : (u | 0x80000000u);
}
__device__ __forceinline__ float dec_f(unsigned u) {
  if (u == 0u) return 0.0f;  // "no edge" sentinel (reference maps non-finite max to 0)
  return (u & 0x80000000u) ? __uint_as_float(u & 0x7fffffffu) : __uint_as_float(~u);
}

// ---------------- generic WMMA f32 GEMM: C = A(MxK,lda) @ B(KxN) + bias -------
// One wave computes a 16 x (NT*16) tile with V_WMMA_F32_16X16X4_F32.
// B is staged through LDS in K-chunks of KC_ rows, stored ROW-PAIR INTERLEAVED:
// element (r,c) lives at Bs[(r>>1)*S + 2c + (r&1)], so each lane's B operand
// (rows k+klo, k+klo+1 of one column) is one aligned 8-byte ds_load_b64 into an
// even VGPR pair -- no repacking movs before the WMMA. S mod 64 == 32 puts the
// two half-wave row groups (klo = 0 vs 2) on disjoint LDS bank halves.
// A streams via global_load_b64. Out-of-range waves compute a duplicate clamped
// tile (never stored) so all waves reach all barriers and EXEC stays all-ones.
template <int NT>
__global__ void gemm_bias_wmma(const float* __restrict__ A, int lda,
                               const float* __restrict__ B,
                               const float* __restrict__ bias,
                               float* __restrict__ C, int ldc,
                               int M, int K, int Ncols) {
  constexpr int W = NT * 16;
  constexpr int S = ((2 * W) % 64 == 32) ? (2 * W) : (2 * W + 32);  // pair-row stride
  extern __shared__ float Bs[];  // (KC_/2) * S floats

  const int lane = threadIdx.x & 31;
  const int wave = threadIdx.x >> 5;
  const int mtile = blockIdx.x * (blockDim.x >> 5) + wave;
  const int row0 = mtile * 16;
  const int colBase = blockIdx.y * W;

  const int m = lane & 15;          // A row within tile
  const int klo = (lane >> 4) * 2;  // K pair 0..1 or 2..3
  int arow = row0 + m;
  if (arow >= M) arow = M - 1;      // clamp; stores are guarded
  const float* Aptr = A + (size_t)arow * lda + klo;

  v8f acc[NT] = {};
  for (int kb = 0; kb < K; kb += KC_) {
    const int kc = (K - kb < KC_) ? (K - kb) : KC_;
    __syncthreads();  // previous chunk fully consumed
    for (int i = threadIdx.x; i < kc * W; i += blockDim.x) {
      int r = i / W, c = i % W;
      int ncol = colBase + c;
      Bs[(r >> 1) * S + 2 * c + (r & 1)] =
          (ncol < Ncols) ? B[(size_t)(kb + r) * Ncols + ncol] : 0.f;
    }
    __syncthreads();

    const float* Bp = Bs + (klo >> 1) * S + (lane & 15) * 2;
#pragma unroll 2
    for (int k = 0; k < kc; k += 4) {
      v2f a = *reinterpret_cast<const v2f*>(Aptr);  // 8B-aligned (lda, klo even)
      Aptr += 4;
      v2f b[NT];
#pragma unroll
      for (int t = 0; t < NT; ++t)
        b[t] = *reinterpret_cast<const v2f*>(Bp + t * 32);  // one ds_load_b64 each
      Bp += 2 * S;
#pragma unroll
      for (int t = 0; t < NT; ++t)
        acc[t] = __builtin_amdgcn_wmma_f32_16x16x4_f32(
            false, a, false, b[t], (short)0, acc[t], false, false);
    }
  }

  const int rowHi = (lane >> 4) * 8;
#pragma unroll
  for (int t = 0; t < NT; ++t) {
    int ncol = colBase + t * 16 + (lane & 15);
    if (ncol >= Ncols) continue;
    float bv = bias ? bias[ncol] : 0.f;
#pragma unroll
    for (int r = 0; r < 8; ++r) {
      int row = row0 + rowHi + r;
      if (row < M) C[(size_t)row * ldc + ncol] = acc[t][r] + bv;
    }
  }
}

// ---------------- pack [Wb | Wdz] into one 128x40 B and 40-wide bias ----------
__global__ void pack_B(const float* __restrict__ Wb, const float* __restrict__ bb,
                       const float* __restrict__ Wdz, const float* __restrict__ bdz,
                       float* __restrict__ Bcat, float* __restrict__ biascat) {
  int i = blockIdx.x * blockDim.x + threadIdx.x;
  if (i < 128 * 40) {
    int r = i / 40, c = i % 40;
    Bcat[i] = (c < 8) ? Wb[r * 8 + c] : Wdz[r * 32 + (c - 8)];
  }
  if (i < 40) biascat[i] = (i < 8) ? bb[i] : bdz[i - 8];
}

// ---------------- per-node prep: k_pts (Wk_v einsum), rotated q_pts, zero accums
__global__ void node_prep(const float* __restrict__ tfn_t, const float* __restrict__ Wk_v,
                          const float* __restrict__ qp_raw, const float* __restrict__ rot,
                          const float* __restrict__ trans,
                          float* __restrict__ k_pts, float* __restrict__ q_pts,
                          float* __restrict__ o_acc, float* __restrict__ opt_acc,
                          float* __restrict__ opair_acc, float* __restrict__ s_acc,
                          unsigned* __restrict__ m_enc, int n_nodes) {
  for (int n = blockIdx.x * blockDim.x + threadIdx.x; n < n_nodes;
       n += gridDim.x * blockDim.x) {
    // k_vec[q][x] = sum_p vecs[p][x] * Wk_v[p][q]
    const float* vrow = tfn_t + (size_t)n * (FS_ + 3 * PIN_) + FS_;
    float kv[PQK_][3];
#pragma unroll
    for (int q = 0; q < PQK_; ++q)
#pragma unroll
      for (int x = 0; x < 3; ++x) {
        float acc = 0.f;
        for (int p = 0; p < PIN_; ++p) acc += vrow[p * 3 + x] * Wk_v[p * PQK_ + q];
        kv[q][x] = acc;
      }
    // reference permutation: k_pts flat index i -> kv[i%8][i/8]
#pragma unroll
    for (int i = 0; i < 24; ++i) k_pts[(size_t)n * 24 + i] = kv[i & 7][i / 8];

    float R[9], tr[3];
#pragma unroll
    for (int i = 0; i < 9; ++i) R[i] = rot[(size_t)n * 9 + i];
#pragma unroll
    for (int i = 0; i < 3; ++i) tr[i] = trans[(size_t)n * 3 + i];
    const float* qr = qp_raw + (size_t)n * 192;
    float* qo = q_pts + (size_t)n * 192;
    for (int p64 = 0; p64 < 64; ++p64) {
      float rx = qr[p64], ry = qr[64 + p64], rz = qr[128 + p64];
#pragma unroll
      for (int i = 0; i < 3; ++i)
        qo[p64 * 3 + i] = R[i * 3 + 0] * rx + R[i * 3 + 1] * ry + R[i * 3 + 2] * rz + tr[i];
    }
    // zero accumulators
    for (int i = 0; i < 128; ++i) o_acc[(size_t)n * 128 + i] = 0.f;
    for (int i = 0; i < 24; ++i) opt_acc[(size_t)n * 24 + i] = 0.f;
    for (int i = 0; i < 256; ++i) opair_acc[(size_t)n * 256 + i] = 0.f;
#pragma unroll
    for (int h = 0; h < 8; ++h) { s_acc[(size_t)n * 8 + h] = 0.f; m_enc[(size_t)n * 8 + h] = 0u; }
  }
}

// ---------------- per-(edge,head) logits + segment max ------------------------
__global__ void edge_logits(const float* __restrict__ q, const float* __restrict__ k,
                            const float* __restrict__ q_pts, const float* __restrict__ k_pts,
                            const float* __restrict__ bz, const int* __restrict__ edge_index,
                            const float* __restrict__ mask,
                            const float* __restrict__ head_weights,
                            float* __restrict__ logits, unsigned* __restrict__ m_enc, int E_) {
  int idx = blockIdx.x * blockDim.x + threadIdx.x;
  if (idx >= E_ * 8) return;
  int e = idx >> 3, h = idx & 7;
  int dst = edge_index[e];        // edge_index[0]
  int src = edge_index[E_ + e];   // edge_index[1]

  const float* qp = q + (size_t)src * HC_ + h * CH_;
  const float* kp = k + (size_t)dst * HC_ + h * CH_;
  float qkd = 0.f;
#pragma unroll
  for (int c = 0; c < CH_; ++c) qkd += qp[c] * kp[c];

  float kx = k_pts[(size_t)dst * 24 + h * 3 + 0];
  float ky = k_pts[(size_t)dst * 24 + h * 3 + 1];
  float kz = k_pts[(size_t)dst * 24 + h * 3 + 2];
  const float* qpt = q_pts + (size_t)src * 192 + h * PQK_ * 3;
  float pts = 0.f;
#pragma unroll
  for (int p = 0; p < PQK_; ++p) {
    float dx = qpt[p * 3 + 0] - kx;
    float dy = qpt[p * 3 + 1] - ky;
    float dz = qpt[p * 3 + 2] - kz;
    pts += dx * dx + dy * dy + dz * dz;
  }
  float hw = log1pf(expf(head_weights[h])) * sqrtf(1.0f / 108.0f); // 3*(PQK*9/2)=108
  float em = 100000.0f * (mask[dst] * mask[src] - 1.0f);
  float a = qkd * sqrtf(1.0f / 48.0f) + sqrtf(1.0f / 3.0f) * bz[(size_t)e * 40 + h]
            - 0.5f * pts * hw + em;
  logits[idx] = a;
  atomicMax(&m_enc[(size_t)src * 8 + h], enc_f(a));
}

// ---------------- one wave per edge: exp weights + atomic numerator sums ------
__global__ void edge_agg(const float* __restrict__ logits, const unsigned* __restrict__ m_enc,
                         const float* __restrict__ k, const float* __restrict__ k_pts,
                         const float* __restrict__ bz, const int* __restrict__ edge_index,
                         float* __restrict__ s_acc, float* __restrict__ o_acc,
                         float* __restrict__ opt_acc, float* __restrict__ opair_acc, int E_) {
  __shared__ float sw[8][8];
  const int lane = threadIdx.x & 31;
  const int wv = threadIdx.x >> 5;
  const int e = blockIdx.x * 8 + wv;
  int src = 0, dst = 0;
  if (e < E_) {
    dst = edge_index[e];
    src = edge_index[E_ + e];
    if (lane < 8) {
      float m = dec_f(m_enc[(size_t)src * 8 + lane]);
      float w = expf(logits[(size_t)e * 8 + lane] - m);
      sw[wv][lane] = w;
      atomicAdd(&s_acc[(size_t)src * 8 + lane], w);
    }
  }
  __syncthreads();
  if (e >= E_) return;
  // o accumulation: 128 cols
#pragma unroll
  for (int i = 0; i < 4; ++i) {
    int col = lane + 32 * i;
    float w = sw[wv][col >> 4];
    atomicAdd(&o_acc[(size_t)src * 128 + col], w * k[(size_t)dst * 128 + col]);
  }
  // o_pt accumulation: 24 values
  if (lane < 24) {
    int h = lane / 3;
    atomicAdd(&opt_acc[(size_t)src * 24 + lane], sw[wv][h] * k_pts[(size_t)dst * 24 + lane]);
  }
  // o_pair accumulation: 256 cols
#pragma unroll
  for (int i = 0; i < 8; ++i) {
    int col = lane + 32 * i;
    float w = sw[wv][col >> 5];
    atomicAdd(&opair_acc[(size_t)src * 256 + col], w * bz[(size_t)e * 40 + 8 + (col & 31)]);
  }
}

// ---------------- normalize + inverse frame + assemble feats (N x 416) -------
__global__ void node_finish(const float* __restrict__ s_acc, const float* __restrict__ o_acc,
                            const float* __restrict__ opt_acc, const float* __restrict__ opair_acc,
                            const float* __restrict__ rot, const float* __restrict__ trans,
                            float* __restrict__ feats, int n_nodes) {
  int n = blockIdx.x * blockDim.x + threadIdx.x;
  if (n >= n_nodes) return;
  float inv[8];
#pragma unroll
  for (int h = 0; h < 8; ++h) inv[h] = 1.0f / (s_acc[(size_t)n * 8 + h] + 1e-16f);
  float* f = feats + (size_t)n * OUT_IN_;
  for (int i = 0; i < 128; ++i) f[i] = o_acc[(size_t)n * 128 + i] * inv[i >> 4];

  float R[9], tr[3];
#pragma unroll
  for (int i = 0; i < 9; ++i) R[i] = rot[(size_t)n * 9 + i];
#pragma unroll
  for (int i = 0; i < 3; ++i) tr[i] = trans[(size_t)n * 3 + i];
#pragma unroll
  for (int h = 0; h < 8; ++h) {
    float g0 = opt_acc[(size_t)n * 24 + h * 3 + 0] * inv[h] - tr[0];
    float g1 = opt_acc[(size_t)n * 24 + h * 3 + 1] * inv[h] - tr[1];
    float g2 = opt_acc[(size_t)n * 24 + h * 3 + 2] * inv[h] - tr[2];
    // out[i] = sum_j rot[j][i] * g[j]  (rot transpose)
    float lx = R[0] * g0 + R[3] * g1 + R[6] * g2;
    float ly = R[1] * g0 + R[4] * g1 + R[7] * g2;
    float lz = R[2] * g0 + R[5] * g1 + R[8] * g2;
    f[128 + h] = lx;
    f[136 + h] = ly;
    f[144 + h] = lz;
    f[152 + h] = sqrtf(lx * lx + ly * ly + lz * lz + 1e-8f);
  }
  for (int i = 0; i < 256; ++i) f[160 + i] = opair_acc[(size_t)n * 256 + i] * inv[i >> 5];
}

// =============================================================================
extern "C" void kernel_launch(void* const* d_in, const int* in_sizes, int n_in,
                              void* d_out, int out_size, void* d_ws, size_t ws_size,
                              hipStream_t stream) {
  const float* frame_s = (const float*)d_in[0];
  const float* tfn_t   = (const float*)d_in[1];
  const float* z       = (const float*)d_in[2];
  const int*   eidx    = (const int*)d_in[3];
  const float* mask    = (const float*)d_in[4];
  const float* rot     = (const float*)d_in[5];
  const float* trans   = (const float*)d_in[6];
  const float* Wq      = (const float*)d_in[7];
  const float* bq      = (const float*)d_in[8];
  const float* Wk_s    = (const float*)d_in[9];
  const float* Wk_v    = (const float*)d_in[10];
  const float* Wqp     = (const float*)d_in[11];
  const float* bqp     = (const float*)d_in[12];
  const float* Wb      = (const float*)d_in[13];
  const float* bb      = (const float*)d_in[14];
  const float* Wdz     = (const float*)d_in[15];
  const float* bdz     = (const float*)d_in[16];
  const float* hwts    = (const float*)d_in[17];
  const float* Wout    = (const float*)d_in[18];
  const float* bout    = (const float*)d_in[19];
  float* out = (float*)d_out;

  const int n = in_sizes[0] / CS;       // 20000
  const int e = in_sizes[2] / CZ;       // 640000

  // workspace layout (floats)
  float* ws = (float*)d_ws;
  size_t off = 0;
  float* q_ws    = ws + off; off += (size_t)n * 128;
  float* qp_raw  = ws + off; off += (size_t)n * 192;
  float* q_pts   = ws + off; off += (size_t)n * 192;
  float* k_ws    = ws + off; off += (size_t)n * 128;
  float* k_pts   = ws + off; off += (size_t)n * 24;
  float* Bcat    = ws + off; off += 128 * 40;
  float* biascat = ws + off; off += 64;
  float* bz      = ws + off; off += (size_t)e * 40;
  float* logits  = ws + off; off += (size_t)e * 8;
  unsigned* m_enc = (unsigned*)(ws + off); off += (size_t)n * 8;
  float* s_acc   = ws + off; off += (size_t)n * 8;
  float* o_acc   = ws + off; off += (size_t)n * 128;
  float* opt_acc = ws + off; off += (size_t)n * 24;
  float* opair   = ws + off; off += (size_t)n * 256;
  float* feats   = ws + off; off += (size_t)n * OUT_IN_;
  (void)ws_size; (void)n_in; (void)out_size;

  const int mt_n = (n + 15) / 16;
  const int gx_n = (mt_n + 3) / 4;     // 4 waves (M-tiles) per 128-thread block
  const int mt_e = (e + 15) / 16;
  const int gx_e = (mt_e + 3) / 4;

  // pair-row-interleaved LDS: (KC_/2) * S floats
  const size_t shm4 = (size_t)(KC_ / 2) * (2 * 64 + 32) * sizeof(float);  // 40960 B
  const size_t shm3 = (size_t)(KC_ / 2) * (2 * 48) * sizeof(float);       // 24576 B

  pack_B<<<(128 * 40 + 255) / 256, 256, 0, stream>>>(Wb, bb, Wdz, bdz, Bcat, biascat);

  // node projections (WMMA f32 16x16x4, B staged in LDS)
  gemm_bias_wmma<4><<<dim3(gx_n, 2), 128, shm4, stream>>>(frame_s, CS, Wq, bq, q_ws, 128, n, CS, 128);
  gemm_bias_wmma<4><<<dim3(gx_n, 3), 128, shm4, stream>>>(frame_s, CS, Wqp, bqp, qp_raw, 192, n, CS, 192);
  gemm_bias_wmma<4><<<dim3(gx_n, 2), 128, shm4, stream>>>(tfn_t, FS_ + 3 * PIN_, Wk_s, nullptr, k_ws, 128, n, FS_, 128);

  node_prep<<<(n + 127) / 128, 128, 0, stream>>>(tfn_t, Wk_v, qp_raw, rot, trans, k_pts, q_pts,
                                                 o_acc, opt_acc, opair, s_acc, m_enc, n);

  // edge GEMM: z @ [Wb|Wdz]; wave tile 16x48 covers all 40 cols -> z streamed once
  gemm_bias_wmma<3><<<dim3(gx_e, 1), 128, shm3, stream>>>(z, CZ, Bcat, biascat, bz, 40, e, CZ, 40);

  edge_logits<<<((size_t)e * 8 + 255) / 256, 256, 0, stream>>>(q_ws, k_ws, q_pts, k_pts, bz,
                                                               eidx, mask, hwts, logits, m_enc, e);

  edge_agg<<<(e + 7) / 8, 256, 0, stream>>>(logits, m_enc, k_ws, k_pts, bz, eidx,
                                            s_acc, o_acc, opt_acc, opair, e);

  node_finish<<<(n + 127) / 128, 128, 0, stream>>>(s_acc, o_acc, opt_acc, opair, rot, trans,
                                                   feats, n);

  // output GEMM: feats @ Wout + bout -> d_out
  gemm_bias_wmma<4><<<dim3(gx_n, 6), 128, shm4, stream>>>(feats, OUT_IN_, Wout, bout, out, CS, n,
                                                          OUT_IN_, CS);
}